// QuantumFilter_65481071406075
// MI455X (gfx1250) — compile-verified
//
#include <hip/hip_runtime.h>
#include <math.h>

// ---------------------------------------------------------------------------
// QuantumFilter on MI455X (gfx1250).
// z_w(sample) = s0^T M_w s0 with M_w = Re(U^H D_w U) a fixed real symmetric
// 16x16 observable (U = the two weight layers), s0 = real RY product state.
// Batched via V_WMMA_F32_16X16X4_F32 chains computing T' = M_w * S^T so that
// each lane's output column IS its own sample (no permute storm).
// ---------------------------------------------------------------------------

#define NQ    4
#define DIM   16            // 2^NQ
#define NPOS  3969          // 63*63 patch positions
#define NB    64            // batch
#define NS    (NB * NPOS)   // 254016 samples
#define IMG_H 128
#define IMG_W 128
#define IMG_STRIDE (3*IMG_H*IMG_W)   // floats per image
#define SPB   256           // samples per block (8 waves x 2 tiles x 16)

typedef float v8f __attribute__((ext_vector_type(8)));
typedef float v2f __attribute__((ext_vector_type(2)));

// --------------------------------------------------------------------------
// Kernel 1: build the 4 observables M_w from weights (one 256-thread block).
// LDS holds U (16x16 complex); thread (i,j) owns element U[i][j].
// Wire w <-> bit (3-w) (wire 0 is MSB of the flattened (2,2,2,2) index).
// --------------------------------------------------------------------------
__global__ __launch_bounds__(256) void build_obs(const float* __restrict__ W,
                                                 float* __restrict__ Mg)
{
    __shared__ float Ur[DIM][DIM];
    __shared__ float Ui[DIM][DIM];
    const int tid = threadIdx.x;
    const int i = tid >> 4, j = tid & 15;
    Ur[i][j] = (i == j) ? 1.f : 0.f;
    Ui[i][j] = 0.f;
    __syncthreads();

    for (int l = 0; l < 2; ++l) {
        for (int q = 0; q < NQ; ++q) {
            const float phi = W[(l*NQ + q)*3 + 0];
            const float th  = W[(l*NQ + q)*3 + 1];
            const float om  = W[(l*NQ + q)*3 + 2];
            const float ct = cosf(0.5f*th), st = sinf(0.5f*th);
            const float ap = 0.5f*(phi + om), am = 0.5f*(phi - om);
            const float cap = cosf(ap), sap = sinf(ap);
            const float cam = cosf(am), sam = sinf(am);
            // m00 = e^{-i ap} c; m01 = -e^{+i am} s; m10 = e^{-i am} s; m11 = e^{+i ap} c
            const float m00r =  ct*cap, m00i = -ct*sap;
            const float m01r = -st*cam, m01i = -st*sam;
            const float m10r =  st*cam, m10i = -st*sam;
            const float m11r =  ct*cap, m11i =  ct*sap;
            const int m  = 1 << (3 - q);
            const int i0 = i & ~m, i1 = i | m;
            const float u0r = Ur[i0][j], u0i = Ui[i0][j];
            const float u1r = Ur[i1][j], u1i = Ui[i1][j];
            __syncthreads();
            float nr, ni;
            if (i & m) {
                nr = m10r*u0r - m10i*u0i + m11r*u1r - m11i*u1i;
                ni = m10r*u0i + m10i*u0r + m11r*u1i + m11i*u1r;
            } else {
                nr = m00r*u0r - m00i*u0i + m01r*u1r - m01i*u1i;
                ni = m00r*u0i + m00i*u0r + m01r*u1i + m01i*u1r;
            }
            Ur[i][j] = nr; Ui[i][j] = ni;
            __syncthreads();
        }
        // CNOT ring (0,1)(1,2)(2,3)(3,0): row permutation of U
        for (int e = 0; e < NQ; ++e) {
            const int cm = 1 << (3 - e), tm = 1 << (3 - ((e + 1) & 3));
            const int src = (i & cm) ? (i ^ tm) : i;
            const float ar = Ur[src][j], ai = Ui[src][j];
            __syncthreads();
            Ur[i][j] = ar; Ui[i][j] = ai;
            __syncthreads();
        }
    }
    // M_w[i][j] = sum_k d_w(k) * Re(conj(U[k][i]) * U[k][j])
    float acc[NQ] = {0.f, 0.f, 0.f, 0.f};
    for (int k = 0; k < DIM; ++k) {
        const float re = Ur[k][i]*Ur[k][j] + Ui[k][i]*Ui[k][j];
#pragma unroll
        for (int w = 0; w < NQ; ++w)
            acc[w] += ((k >> (3 - w)) & 1) ? -re : re;
    }
#pragma unroll
    for (int w = 0; w < NQ; ++w)
        Mg[w*256 + i*16 + j] = acc[w];
}

// --------------------------------------------------------------------------
// Kernel 2: per patch-position max over (batch=64, ch=3, 4x4 window).
// One thread per batch image; base address hoisted, window walked with
// constant-stride float2 loads (col0 is even -> 8B aligned).  Inner body is
// just load + v_max_f32.  scale[p] = (max > 0) ? pi/max : 0.
// --------------------------------------------------------------------------
__global__ __launch_bounds__(64) void patch_scale(const float* __restrict__ x,
                                                  float* __restrict__ scale)
{
    const int pos  = blockIdx.x;
    const int row0 = 2 * (pos / 63), col0 = 2 * (pos % 63);
    const float* p0 = x + (size_t)threadIdx.x * IMG_STRIDE + row0 * IMG_W + col0;

    float mx = -3.402823466e38f;
#pragma unroll
    for (int ch = 0; ch < 3; ++ch) {
        const float* pc = p0 + ch * (IMG_H * IMG_W);
#pragma unroll
        for (int pr = 0; pr < 4; ++pr) {
            const float2 u = *(const float2*)(pc + pr * IMG_W);
            const float2 v = *(const float2*)(pc + pr * IMG_W + 2);
            mx = fmaxf(mx, fmaxf(fmaxf(u.x, u.y), fmaxf(v.x, v.y)));
        }
    }
#pragma unroll
    for (int off = 16; off > 0; off >>= 1)
        mx = fmaxf(mx, __shfl_xor(mx, off, 32));
    __shared__ float wmax[2];
    if ((threadIdx.x & 31) == 0) wmax[threadIdx.x >> 5] = mx;
    __syncthreads();
    if (threadIdx.x == 0) {
        const float m = fmaxf(wmax[0], wmax[1]);
        scale[pos] = (m > 0.f) ? (3.14159265358979323846f / m) : 0.f;
    }
}

// --------------------------------------------------------------------------
// Kernel 3: 256 samples/block, thread t owns sample sBase+t.
// Phase A: build the 16-amplitude product state row into LDS.
// Phase B: per wave, two 16-sample tiles.  For each observable w:
//   T' = M_w * S^T  via 4 chained V_WMMA_F32_16X16X4_F32
//     A = M_w  (element (M=ln,K)  -> Ml[w][K][ln], symmetric M -> same loads)
//     B = S^T  (element (K,N=ln)  -> S[rowbase+ln][K])
//   lane l then holds T'[j=r+8*half][sample=ln] in acc[r]; dot with the
//   sample's own amplitudes (LDS column read) + ONE shfl_xor(16) join.
//   The lane holding the finished z for (half==sub) is the sample's owner
//   thread -> direct per-thread float4 store, no staging, no exec juggling.
// --------------------------------------------------------------------------
__global__ __launch_bounds__(256) void qsim_wmma(const float* __restrict__ x,
                                                 const float* __restrict__ scale,
                                                 const float* __restrict__ Mg,
                                                 float* __restrict__ out)
{
    __shared__ float S[SPB][17];        // +1 pad: conflict-free column reads
    __shared__ float Ml[4][16][16];

    const int t = threadIdx.x;
    const int sBase = blockIdx.x * SPB;
    const int s = sBase + t;

    for (int i2 = t; i2 < 4*256; i2 += 256) ((float*)Ml)[i2] = Mg[i2];

    // ---- Phase A: angles -> product state s0 (invalid samples -> |0..0>) ----
    float a0 = 0.f, a1 = 0.f, a2 = 0.f, a3 = 0.f;
    if (s < NS) {
        const int b = s / NPOS, p = s % NPOS;
        const int r = p / 63, c = p % 63;
        const float sc = scale[p];                        // pi/max (or 0)
        const float* px = x + (size_t)b * IMG_STRIDE + (2*r) * IMG_W + 2*c;
        a0 = sc*px[0]; a1 = sc*px[1]; a2 = sc*px[2]; a3 = sc*px[3];
    }
    // half-angles are in [0, pi/2]: hardware sin/cos is plenty accurate here
    float cs[4], sn[4];
    __sincosf(0.5f*a0, &sn[0], &cs[0]);
    __sincosf(0.5f*a1, &sn[1], &cs[1]);
    __sincosf(0.5f*a2, &sn[2], &cs[2]);
    __sincosf(0.5f*a3, &sn[3], &cs[3]);
#pragma unroll
    for (int j = 0; j < 16; ++j)
        S[t][j] = (j & 8 ? sn[0] : cs[0]) * (j & 4 ? sn[1] : cs[1])
                * (j & 2 ? sn[2] : cs[2]) * (j & 1 ? sn[3] : cs[3]);
    __syncthreads();

    // ---- Phase B (uniform control flow around WMMA, EXEC all 1s) ----
    const int lane = t & 31;
    const int half = lane >> 4;     // 0: lanes 0-15, 1: lanes 16-31
    const int ln   = lane & 15;
    const int wv   = t >> 5;

    // A operand = M_w, hoisted once per wave, reused for both sub-tiles.
    // f32 16x4 A layout: lane -> M=ln; VGPR0 -> K=k0, VGPR1 -> K=k0+1,
    // k0 = kk*4 + 2*half.  M symmetric: M_w[ln][k] == Ml[w][k][ln].
    v2f Areg[4][4];
#pragma unroll
    for (int w = 0; w < 4; ++w)
#pragma unroll
        for (int kk = 0; kk < 4; ++kk) {
            const int k0 = kk*4 + 2*half;
            Areg[w][kk].x = Ml[w][k0][ln];
            Areg[w][kk].y = Ml[w][k0 + 1][ln];
        }

    for (int sub = 0; sub < 2; ++sub) {
        const int rowbase = wv*32 + sub*16;

        // B operand = S^T chunk: element (K=k0, N=ln) -> S[rowbase+ln][k0]
        v2f Breg[4];
#pragma unroll
        for (int kk = 0; kk < 4; ++kk) {
            const int k0 = kk*4 + 2*half;
            Breg[kk].x = S[rowbase + ln][k0];
            Breg[kk].y = S[rowbase + ln][k0 + 1];
        }
        // This lane's dot operand: amplitudes j = 8*half + r of sample ln
        float Scol[8];
#pragma unroll
        for (int rr = 0; rr < 8; ++rr)
            Scol[rr] = S[rowbase + ln][rr + 8*half];

        float z[4];
#pragma unroll
        for (int w = 0; w < 4; ++w) {
            v8f acc = {0.f,0.f,0.f,0.f,0.f,0.f,0.f,0.f};
#pragma unroll
            for (int kk = 0; kk < 4; ++kk)
                acc = __builtin_amdgcn_wmma_f32_16x16x4_f32(
                        false, Areg[w][kk], false, Breg[kk], (short)0, acc,
                        false, false);
            // acc[r] = T'[j = r+8*half][sample = ln]; partial dot over this
            // lane's 8 j's, then join the two halves with one xor-16 shuffle.
            float pv = Scol[0]*acc[0];
#pragma unroll
            for (int rr = 1; rr < 8; ++rr) pv += Scol[rr]*acc[rr];
            z[w] = pv + __shfl_xor(pv, 16);
        }

        // Finished z for sample rowbase+ln sits in lanes with half==sub,
        // which is exactly that sample's owner thread (t == rowbase+ln)
        if (half == sub && s < NS) {
            float4 z4 = make_float4(z[0], z[1], z[2], z[3]);
            ((float4*)out)[s] = z4;    // out[4*s + w], coalesced per tile
        }
    }
}

// --------------------------------------------------------------------------
extern "C" void kernel_launch(void* const* d_in, const int* in_sizes, int n_in,
                              void* d_out, int out_size, void* d_ws, size_t ws_size,
                              hipStream_t stream)
{
    const float* x = (const float*)d_in[0];   // (64,3,128,128) f32
    const float* W = (const float*)d_in[1];   // (2,4,3) f32
    float* out     = (float*)d_out;           // (64,15876) f32

    float* Mg    = (float*)d_ws;              // 4*256 floats  (4 KB)
    float* scale = Mg + 4*256;                // 3969 floats   (~16 KB)

    build_obs  <<<1,    256, 0, stream>>>(W, Mg);
    patch_scale<<<NPOS, 64, 0, stream>>>(x, scale);
    const int blocks = (NS + SPB - 1) / SPB;  // 993
    qsim_wmma  <<<blocks, SPB, 0, stream>>>(x, scale, Mg, out);
}